// HierarchicalCoordinationModule_78314433675524
// MI455X (gfx1250) — compile-verified
//

#include <hip/hip_runtime.h>
#include <hip/hip_bf16.h>
#include <math.h>

// ---------------------------------------------------------------------------
// HierarchicalCoordinationModule for MI455X (gfx1250, wave32).
// All dense math on V_WMMA_F32_16X16X4_F32 (f32 in/out matrix core): tensors
// are fp32 and the workload is HBM-bound, so no downcast is warranted.
// Attention is fused (scores -> softmax -> attn-write -> ctx) per 16-query
// row block so the 268MB attn tensor is written exactly once and never
// re-read from HBM; K/V slices (512KB each/head) stay L2-resident across re-reads.
// ---------------------------------------------------------------------------

#define N_AGENTS 4096
#define S_DIM    64
#define HID      256
#define DM       128
#define NH       4
#define DK       32
#define A_DIM    32
#define EPS      1e-5f
#define INV_SCALE 0.17677669529663687f   // 1/sqrt(32)

typedef __attribute__((ext_vector_type(2))) float v2f;
typedef __attribute__((ext_vector_type(8))) float v8f;

// ---------------------------------------------------------------------------
// Generic f32 WMMA GEMM: C = act(alpha * A@B + bias [+ residual])
// A: MxK row-major (lda), B: KxN row-major (ldb), C: MxN row-major (ldc).
// One wave per 16x16 output tile; K stepped by 4 per WMMA.
// f32 16x16x4 VGPR layouts per CDNA5 ISA 7.12.2.
// ---------------------------------------------------------------------------
__global__ __launch_bounds__(32) void wmma_gemm_f32(
    const float* __restrict__ A, const float* __restrict__ B,
    const float* __restrict__ bias, const float* __restrict__ residual,
    float* __restrict__ C, int K, int lda, int ldb, int ldc,
    float alpha, int act)
{
    const int lane = threadIdx.x & 31;
    const int half = lane >> 4;        // lane group: K pair offset
    const int l    = lane & 15;
    const int koff = half * 2;
    const int m0 = blockIdx.y * 16;
    const int n0 = blockIdx.x * 16;

    const float* __restrict__ arow = A + (size_t)(m0 + l) * lda;

    v8f acc = {};
    for (int k = 0; k < K; k += 4) {
        v2f a, b;
        a.x = arow[k + koff];
        a.y = arow[k + koff + 1];
        b.x = B[(size_t)(k + koff)     * ldb + n0 + l];
        b.y = B[(size_t)(k + koff + 1) * ldb + n0 + l];
        acc = __builtin_amdgcn_wmma_f32_16x16x4_f32(
            false, a, false, b, (short)0, acc, false, false);
    }

    const int n = n0 + l;
    float bv = bias ? bias[n] : 0.0f;
#pragma unroll
    for (int i = 0; i < 8; ++i) {
        const int m = m0 + i + half * 8;
        float v = acc[i] * alpha + bv;
        if (residual) v += residual[(size_t)m * ldc + n];
        if (act) v = fmaxf(v, 0.0f);
        C[(size_t)m * ldc + n] = v;
    }
}

// ---------------------------------------------------------------------------
// Fused attention: one wave owns 16 queries of one head.
// Pass 1: WMMA scores over all key tiles -> row max (shfl_xor reduce).
// Pass 2: recompute scores -> row exp-sum.
// Pass 3: recompute scores -> write normalized probs to attn (the only HBM
//         write of the 268MB tensor) and accumulate ctx = P@V via a 16x16
//         LDS transpose of P (C-layout -> A-layout) + WMMA.
// Score recompute is ~free (K=32) vs. saving two full HBM passes over attn.
// ---------------------------------------------------------------------------
__global__ __launch_bounds__(32) void fused_attention_wmma(
    const float* __restrict__ q, const float* __restrict__ k,
    const float* __restrict__ v, float* __restrict__ attn,
    float* __restrict__ ctx)
{
    __shared__ float ptile[16][17];          // P transpose staging (pad: no bank conflict)

    const int lane = threadIdx.x & 31;
    const int half = lane >> 4;
    const int l    = lane & 15;
    const int koff = half * 2;
    const int h  = blockIdx.y;
    const int m0 = blockIdx.x * 16;

    // Preload Q fragments for all 8 K-steps (16x32 slice lives in registers).
    const float* __restrict__ qrow = q + (size_t)(m0 + l) * DM + h * DK;
    v2f aq[8];
#pragma unroll
    for (int s = 0; s < 8; ++s) {
        aq[s].x = qrow[4 * s + koff];
        aq[s].y = qrow[4 * s + koff + 1];
    }

    const float* __restrict__ kh = k + h * DK;
    const float* __restrict__ vh = v + h * DK;
    float* __restrict__ arow = attn + (size_t)h * N_AGENTS * N_AGENTS;

    // ---- pass 1: row max ----
    float mx[8];
#pragma unroll
    for (int i = 0; i < 8; ++i) mx[i] = -INFINITY;
    for (int kb = 0; kb < N_AGENTS; kb += 16) {
        const float* __restrict__ krow = kh + (size_t)(kb + l) * DM;
        v8f acc = {};
#pragma unroll
        for (int s = 0; s < 8; ++s) {
            v2f b; b.x = krow[4 * s + koff]; b.y = krow[4 * s + koff + 1];
            acc = __builtin_amdgcn_wmma_f32_16x16x4_f32(
                false, aq[s], false, b, (short)0, acc, false, false);
        }
#pragma unroll
        for (int i = 0; i < 8; ++i) mx[i] = fmaxf(mx[i], acc[i] * INV_SCALE);
    }
#pragma unroll
    for (int off = 8; off >= 1; off >>= 1)
#pragma unroll
        for (int i = 0; i < 8; ++i) mx[i] = fmaxf(mx[i], __shfl_xor(mx[i], off, 32));

    // ---- pass 2: row exp-sum ----
    float sum[8];
#pragma unroll
    for (int i = 0; i < 8; ++i) sum[i] = 0.0f;
    for (int kb = 0; kb < N_AGENTS; kb += 16) {
        const float* __restrict__ krow = kh + (size_t)(kb + l) * DM;
        v8f acc = {};
#pragma unroll
        for (int s = 0; s < 8; ++s) {
            v2f b; b.x = krow[4 * s + koff]; b.y = krow[4 * s + koff + 1];
            acc = __builtin_amdgcn_wmma_f32_16x16x4_f32(
                false, aq[s], false, b, (short)0, acc, false, false);
        }
#pragma unroll
        for (int i = 0; i < 8; ++i) sum[i] += __expf(acc[i] * INV_SCALE - mx[i]);
    }
#pragma unroll
    for (int off = 8; off >= 1; off >>= 1)
#pragma unroll
        for (int i = 0; i < 8; ++i) sum[i] += __shfl_xor(sum[i], off, 32);
    float inv[8];
#pragma unroll
    for (int i = 0; i < 8; ++i) inv[i] = 1.0f / sum[i];

    // ---- pass 3: write probs once + accumulate ctx = P @ V ----
    v8f cacc0 = {}, cacc1 = {};
    for (int kb = 0; kb < N_AGENTS; kb += 16) {
        const float* __restrict__ krow = kh + (size_t)(kb + l) * DM;
        v8f acc = {};
#pragma unroll
        for (int s = 0; s < 8; ++s) {
            v2f b; b.x = krow[4 * s + koff]; b.y = krow[4 * s + koff + 1];
            acc = __builtin_amdgcn_wmma_f32_16x16x4_f32(
                false, aq[s], false, b, (short)0, acc, false, false);
        }
#pragma unroll
        for (int i = 0; i < 8; ++i) {
            const float p = __expf(acc[i] * INV_SCALE - mx[i]) * inv[i];
            arow[(size_t)(m0 + i + half * 8) * N_AGENTS + kb + l] = p;   // attn out
            ptile[i + half * 8][l] = p;                                  // for transpose
        }
        __syncthreads();   // single-wave WG: S_NOP, but orders LDS for clarity
        // ctx tile: A = P (16x16 from LDS), B = V slice (16 keys x 32 dims)
#pragma unroll
        for (int s2 = 0; s2 < 4; ++s2) {
            v2f a;
            a.x = ptile[l][4 * s2 + koff];
            a.y = ptile[l][4 * s2 + koff + 1];
            const float* __restrict__ v0 = vh + (size_t)(kb + 4 * s2 + koff) * DM;
            const float* __restrict__ v1 = v0 + DM;
            v2f b0, b1;
            b0.x = v0[l];      b0.y = v1[l];
            b1.x = v0[16 + l]; b1.y = v1[16 + l];
            cacc0 = __builtin_amdgcn_wmma_f32_16x16x4_f32(
                false, a, false, b0, (short)0, cacc0, false, false);
            cacc1 = __builtin_amdgcn_wmma_f32_16x16x4_f32(
                false, a, false, b1, (short)0, cacc1, false, false);
        }
        __syncthreads();
    }
#pragma unroll
    for (int i = 0; i < 8; ++i) {
        float* __restrict__ crow = ctx + (size_t)(m0 + i + half * 8) * DM + h * DK;
        crow[l]      = cacc0[i];
        crow[16 + l] = cacc1[i];
    }
}

// ---------------------------------------------------------------------------
// Row softmax (in place) — used for the 32-wide task head only.
// ---------------------------------------------------------------------------
__global__ void softmax_rows(float* __restrict__ data, int cols)
{
    __shared__ float buf[4096];
    __shared__ float red[256];
    const int t  = threadIdx.x;
    const int nt = blockDim.x;
    float* __restrict__ p = data + (size_t)blockIdx.x * cols;

    float mx = -INFINITY;
    for (int c = t; c < cols; c += nt) { float v = p[c]; buf[c] = v; mx = fmaxf(mx, v); }
    red[t] = mx; __syncthreads();
    for (int s = nt >> 1; s > 0; s >>= 1) { if (t < s) red[t] = fmaxf(red[t], red[t + s]); __syncthreads(); }
    mx = red[0]; __syncthreads();

    float sum = 0.0f;
    for (int c = t; c < cols; c += nt) { float e = __expf(buf[c] - mx); buf[c] = e; sum += e; }
    red[t] = sum; __syncthreads();
    for (int s = nt >> 1; s > 0; s >>= 1) { if (t < s) red[t] += red[t + s]; __syncthreads(); }
    const float inv = 1.0f / red[0];
    __syncthreads();
    for (int c = t; c < cols; c += nt) p[c] = buf[c] * inv;
}

// ---------------------------------------------------------------------------
// LayerNorm over DM=128 (one block of 128 threads per row; biased variance).
// ---------------------------------------------------------------------------
__global__ __launch_bounds__(128) void layernorm_rows(
    const float* __restrict__ x, const float* __restrict__ g,
    const float* __restrict__ b, float* __restrict__ y)
{
    __shared__ float red[128];
    const int t = threadIdx.x;
    const float* __restrict__ p = x + (size_t)blockIdx.x * DM;
    const float v = p[t];

    red[t] = v; __syncthreads();
    for (int s = 64; s > 0; s >>= 1) { if (t < s) red[t] += red[t + s]; __syncthreads(); }
    const float mu = red[0] * (1.0f / DM); __syncthreads();

    const float d = v - mu;
    red[t] = d * d; __syncthreads();
    for (int s = 64; s > 0; s >>= 1) { if (t < s) red[t] += red[t + s]; __syncthreads(); }
    const float var = red[0] * (1.0f / DM);

    y[(size_t)blockIdx.x * DM + t] = d * rsqrtf(var + EPS) * g[t] + b[t];
}

// ---------------------------------------------------------------------------
// Strategic mat-vec h1[j] += sum_r gs[r]*sw1[r,j]  (268MB streamed, HBM-bound)
// 512 blocks x 512 rows each; gs chunk staged in LDS; coalesced sw1 reads
// with prefetch hints (global_prefetch_b8).
// ---------------------------------------------------------------------------
__global__ __launch_bounds__(256) void strat_phase1(
    const float* __restrict__ gs, const float* __restrict__ sw1,
    float* __restrict__ h1acc)
{
    __shared__ float gsl[512];
    const int t  = threadIdx.x;                 // column 0..255
    const int r0 = blockIdx.x * 512;
    gsl[t]       = gs[r0 + t];
    gsl[t + 256] = gs[r0 + t + 256];
    __syncthreads();

    float sum = 0.0f;
    const float* __restrict__ w = sw1 + (size_t)r0 * HID + t;
    for (int r = 0; r < 512; ++r) {
        if ((r & 15) == 0 && r + 16 < 512)
            __builtin_prefetch(w + (size_t)(r + 16) * HID, 0, 0);
        sum += gsl[r] * w[(size_t)r * HID];
    }
    atomicAdd(&h1acc[t], sum);
}

// h1 -> relu(+sb1) -> sw2 -> relu -> sow -> gobj   (tiny: one block)
__global__ __launch_bounds__(256) void strat_phase2(
    const float* __restrict__ h1acc, const float* __restrict__ sb1,
    const float* __restrict__ sw2,  const float* __restrict__ sb2,
    const float* __restrict__ sow,  const float* __restrict__ sob,
    float* __restrict__ gobj)
{
    __shared__ float h1[HID];
    __shared__ float h2[HID];
    const int t = threadIdx.x;
    h1[t] = fmaxf(h1acc[t] + sb1[t], 0.0f);
    __syncthreads();
    float s = sb2[t];
    for (int i = 0; i < HID; ++i) s += h1[i] * sw2[(size_t)i * HID + t];
    h2[t] = fmaxf(s, 0.0f);
    __syncthreads();
    if (t < 64) {
        float g = sob[t];
        for (int i = 0; i < HID; ++i) g += h2[i] * sow[(size_t)i * 64 + t];
        gobj[t] = g;
    }
}

__global__ void zero_buf(float* __restrict__ p, int n)
{
    int i = blockIdx.x * blockDim.x + threadIdx.x;
    if (i < n) p[i] = 0.0f;
}

// tact_in[n, 0:64]=agent_states, [64:128]=gobj broadcast
__global__ void build_tact_in(const float* __restrict__ as,
                              const float* __restrict__ gobj,
                              float* __restrict__ out)
{
    int i = blockIdx.x * blockDim.x + threadIdx.x;   // over 4096*128
    int row = i >> 7, col = i & 127;
    out[i] = (col < 64) ? as[row * 64 + col] : gobj[col - 64];
}

// op_in[n, 0:64]=agent_states, [64:192]=coord, [192:224]=task
__global__ void build_op_in(const float* __restrict__ as,
                            const float* __restrict__ coord,
                            const float* __restrict__ task,
                            float* __restrict__ out)
{
    int i = blockIdx.x * blockDim.x + threadIdx.x;   // over 4096*224
    int row = i / 224, col = i % 224;
    float v;
    if (col < 64)       v = as[row * 64 + col];
    else if (col < 192) v = coord[row * DM + (col - 64)];
    else                v = task[row * 32 + (col - 192)];
    out[i] = v;
}

// values[row] = opf[row] . vw + vb   (N=1 -> plain reduction, not WMMA)
__global__ __launch_bounds__(128) void values_kernel(
    const float* __restrict__ opf, const float* __restrict__ vw,
    const float* __restrict__ vb, float* __restrict__ values)
{
    __shared__ float red[128];
    const int t = threadIdx.x;
    red[t] = opf[(size_t)blockIdx.x * DM + t] * vw[t];
    __syncthreads();
    for (int s = 64; s > 0; s >>= 1) { if (t < s) red[t] += red[t + s]; __syncthreads(); }
    if (t == 0) values[blockIdx.x] = red[0] + vb[0];
}

// ---------------------------------------------------------------------------
extern "C" void kernel_launch(void* const* d_in, const int* in_sizes, int n_in,
                              void* d_out, int out_size, void* d_ws, size_t ws_size,
                              hipStream_t stream)
{
    const float* AS   = (const float*)d_in[0];
    const float* sw1  = (const float*)d_in[1];
    const float* sb1  = (const float*)d_in[2];
    const float* sw2  = (const float*)d_in[3];
    const float* sb2  = (const float*)d_in[4];
    const float* sow  = (const float*)d_in[5];
    const float* sob  = (const float*)d_in[6];
    const float* tw   = (const float*)d_in[7];
    const float* tb   = (const float*)d_in[8];
    const float* wq   = (const float*)d_in[9];
    const float* bq   = (const float*)d_in[10];
    const float* wk   = (const float*)d_in[11];
    const float* bk   = (const float*)d_in[12];
    const float* wv   = (const float*)d_in[13];
    const float* bv   = (const float*)d_in[14];
    const float* wo   = (const float*)d_in[15];
    const float* bo   = (const float*)d_in[16];
    const float* ln_g = (const float*)d_in[17];
    const float* ln_b = (const float*)d_in[18];
    const float* taw1 = (const float*)d_in[19];
    const float* tab1 = (const float*)d_in[20];
    const float* taw2 = (const float*)d_in[21];
    const float* tab2 = (const float*)d_in[22];
    const float* ow1  = (const float*)d_in[23];
    const float* ob1  = (const float*)d_in[24];
    const float* ow2  = (const float*)d_in[25];
    const float* ob2  = (const float*)d_in[26];
    const float* aw   = (const float*)d_in[27];
    const float* ab   = (const float*)d_in[28];
    const float* vw   = (const float*)d_in[29];
    const float* vb   = (const float*)d_in[30];
    const float* cw   = (const float*)d_in[31];
    const float* cb   = (const float*)d_in[32];
    const float* cdw  = (const float*)d_in[33];
    const float* cdb  = (const float*)d_in[34];

    // d_out layout (return order): actions, values, messages, task, attn, gobj
    float* out       = (float*)d_out;
    float* actions_o = out;                         // 4096*32
    float* values_o  = out + 131072;                // 4096
    float* msgs_o    = out + 135168;                // 4096*32
    float* task_o    = out + 266240;                // 4096*32
    float* attn_o    = out + 397312;                // 4*4096*4096
    float* gobj_o    = out + 67506176;              // 64

    // workspace layout (floats)
    float* ws      = (float*)d_ws;
    float* tact_in = ws;                  // 4096*128
    float* tact    = ws + 524288;         // 4096*128
    float* q       = ws + 1048576;        // 4096*128
    float* k       = ws + 1572864;        // 4096*128
    float* v       = ws + 2097152;        // 4096*128
    float* ctx     = ws + 2621440;        // 4096*128
    float* res     = ws + 3145728;        // 4096*128
    float* coord   = ws + 3670016;        // 4096*128
    float* t1      = ws + 4194304;        // 4096*64
    float* op_in   = ws + 4456448;        // 4096*224
    float* opf1    = ws + 5373952;        // 4096*256
    float* opf     = ws + 6422528;        // 4096*128
    float* m1      = ws + 6946816;        // 4096*64
    float* h1acc   = ws + 7208960;        // 256

    // ---- strategic level (HBM-bound 268MB mat-vec) ----
    zero_buf<<<1, 256, 0, stream>>>(h1acc, HID);
    strat_phase1<<<512, 256, 0, stream>>>(AS, sw1, h1acc);
    strat_phase2<<<1, 256, 0, stream>>>(h1acc, sb1, sw2, sb2, sow, sob, gobj_o);

    // ---- tactical encode ----
    build_tact_in<<<(N_AGENTS * DM) / 256, 256, 0, stream>>>(AS, gobj_o, tact_in);
    wmma_gemm_f32<<<dim3(DM / 16, N_AGENTS / 16), 32, 0, stream>>>(
        tact_in, tw, tb, nullptr, tact, DM, DM, DM, DM, 1.0f, 1);

    // ---- Q, K, V projections ----
    wmma_gemm_f32<<<dim3(DM / 16, N_AGENTS / 16), 32, 0, stream>>>(
        tact, wq, bq, nullptr, q, DM, DM, DM, DM, 1.0f, 0);
    wmma_gemm_f32<<<dim3(DM / 16, N_AGENTS / 16), 32, 0, stream>>>(
        tact, wk, bk, nullptr, k, DM, DM, DM, DM, 1.0f, 0);
    wmma_gemm_f32<<<dim3(DM / 16, N_AGENTS / 16), 32, 0, stream>>>(
        tact, wv, bv, nullptr, v, DM, DM, DM, DM, 1.0f, 0);

    // ---- fused attention: attn written once, ctx produced in-pass ----
    fused_attention_wmma<<<dim3(N_AGENTS / 16, NH), 32, 0, stream>>>(
        q, k, v, attn_o, ctx);

    // ---- output proj + residual + layernorm ----
    wmma_gemm_f32<<<dim3(DM / 16, N_AGENTS / 16), 32, 0, stream>>>(
        ctx, wo, bo, tact, res, DM, DM, DM, DM, 1.0f, 0);
    layernorm_rows<<<N_AGENTS, 128, 0, stream>>>(res, ln_g, ln_b, coord);

    // ---- task allocation ----
    wmma_gemm_f32<<<dim3(64 / 16, N_AGENTS / 16), 32, 0, stream>>>(
        coord, taw1, tab1, nullptr, t1, DM, DM, 64, 64, 1.0f, 1);
    wmma_gemm_f32<<<dim3(32 / 16, N_AGENTS / 16), 32, 0, stream>>>(
        t1, taw2, tab2, nullptr, task_o, 64, 64, 32, 32, 1.0f, 0);
    softmax_rows<<<N_AGENTS, 32, 0, stream>>>(task_o, 32);

    // ---- operational level ----
    build_op_in<<<(N_AGENTS * 224) / 256, 256, 0, stream>>>(AS, coord, task_o, op_in);
    wmma_gemm_f32<<<dim3(256 / 16, N_AGENTS / 16), 32, 0, stream>>>(
        op_in, ow1, ob1, nullptr, opf1, 224, 224, 256, 256, 1.0f, 1);
    wmma_gemm_f32<<<dim3(DM / 16, N_AGENTS / 16), 32, 0, stream>>>(
        opf1, ow2, ob2, nullptr, opf, 256, 256, DM, DM, 1.0f, 1);
    wmma_gemm_f32<<<dim3(A_DIM / 16, N_AGENTS / 16), 32, 0, stream>>>(
        opf, aw, ab, nullptr, actions_o, DM, DM, A_DIM, A_DIM, 1.0f, 0);
    values_kernel<<<N_AGENTS, 128, 0, stream>>>(opf, vw, vb, values_o);

    // ---- communication messages ----
    wmma_gemm_f32<<<dim3(64 / 16, N_AGENTS / 16), 32, 0, stream>>>(
        AS, cw, cb, nullptr, m1, 64, 64, 64, 64, 1.0f, 0);
    wmma_gemm_f32<<<dim3(32 / 16, N_AGENTS / 16), 32, 0, stream>>>(
        m1, cdw, cdb, nullptr, msgs_o, 64, 64, 32, 32, 1.0f, 0);
}